// ObjectCondensation_26070451487262
// MI455X (gfx1250) — compile-verified
//
#include <hip/hip_runtime.h>
#include <math.h>

typedef float v2f __attribute__((ext_vector_type(2)));
typedef float v8f __attribute__((ext_vector_type(8)));

#define NOBJ_PAD 544      // 513 objects padded to 34*16; zero-padded entries contribute nothing
#define OC_EPS 0.001f
#define OC_QMIN 1.0f

// ---------------- kernel 0: init scratch (ws is poisoned 0xAA by harness) ----------------
__global__ void oc_init_kernel(unsigned long long* packed, int* nin, int* nnoise) {
    int i = blockIdx.x * blockDim.x + threadIdx.x;
    if (i < NOBJ_PAD) {
        packed[i] = 0xFFFFFFFFull;   // pack(b=0.0, index=0): matches argmax-of-zeros == 0
        nin[i] = 0;
    }
    if (i == 0) *nnoise = 0;
}

// ---------------- kernel 1: per-vertex scan (argmax beta per object, counts) ----------------
__global__ void oc_scan_kernel(const int* __restrict__ vids, const float* __restrict__ beta,
                               int N, unsigned long long* packed, int* nin, int* nnoise) {
    int i = blockIdx.x * blockDim.x + threadIdx.x;
    if (i >= N) return;
    int vid = vids[i];
    int k = vid + 1;                         // object -1 -> column 0
    float b = beta[i];                       // b in [0,1): positive float bits are order-preserving
    unsigned long long key =
        ((unsigned long long)__float_as_uint(b) << 32) |
        (unsigned long long)(0xFFFFFFFFu - (unsigned int)i);   // smaller index wins ties
    atomicMax(&packed[k], key);
    atomicAdd(&nin[k], 1);
    if (vid == -1) atomicAdd(nnoise, 1);
}

// ---------------- kernel 2: per-object condensation data + L_beta (one block, 1024 thr) ----
__global__ void oc_cond_kernel(const unsigned long long* __restrict__ packed,
                               const int* __restrict__ nin,
                               const float* __restrict__ beta,
                               const float* __restrict__ ccoords,
                               const int* __restrict__ pK, int N,
                               float* __restrict__ obj, float* __restrict__ pLbeta) {
    __shared__ float red[1024];
    const int k = threadIdx.x;
    const int Kv = *pK;
    float lb = 0.f;
    if (k < NOBJ_PAD) {
        float xkx = 0.f, xky = 0.f, xkz = 0.f, nk = 0.f, satt = 0.f, srep = 0.f;
        if (k <= Kv) {
            unsigned long long p = packed[k];
            unsigned int idx = 0xFFFFFFFFu - (unsigned int)(p & 0xFFFFFFFFull);
            float bk = beta[idx];
            float t = tanhf(bk);
            float qk = t * t + OC_QMIN;
            float x = ccoords[3 * idx + 0];
            float y = ccoords[3 * idx + 1];
            float z = ccoords[3 * idx + 2];
            float fin = (float)nin[k];
            float fout = (float)N - fin;
            float denom = (float)Kv;         // (K+1) - 1
            satt = qk / ((fin + OC_EPS) * denom);
            srep = qk / ((fout + OC_EPS) * denom);
            xkx = x; xky = y; xkz = z;
            nk = x * x + y * y + z * z;
            lb = (1.f - bk) / denom;
        }
        obj[k]          = xkx;
        obj[544 + k]    = xky;
        obj[1088 + k]   = xkz;
        obj[1632 + k]   = nk;
        obj[2176 + k]   = satt;
        obj[2720 + k]   = srep;
    }
    red[threadIdx.x] = lb;
    __syncthreads();
    for (int s = 512; s > 0; s >>= 1) {
        if (threadIdx.x < s) red[threadIdx.x] += red[threadIdx.x + s];
        __syncthreads();
    }
    if (threadIdx.x == 0) *pLbeta = red[0];
}

// ---------------- kernel 3: main O(N*(K+1)) pass; WMMA f32 16x16x4 for d^2 tiles ----------
// Per wave: one 16-vertex tile. A-row m = [xk, ||xk||^2], B-col i = [-2*xi, 1],
// C preloaded with ||xi||^2 per column => D[m][i] = d^2 directly out of the WMMA.
// Inner loop is a pure repulsive sum; the single member (attractive) element per column
// is corrected once after the loop using the exact coordinate-difference d^2.
__global__ void __launch_bounds__(256)
oc_main_kernel(const float* __restrict__ ccoords, const float* __restrict__ beta,
               const int* __restrict__ vids, const int* __restrict__ pK,
               const float* __restrict__ obj, int N, float* __restrict__ partials) {
    __shared__ __align__(16) float sXKX[NOBJ_PAD];
    __shared__ __align__(16) float sXKY[NOBJ_PAD];
    __shared__ __align__(16) float sXKZ[NOBJ_PAD];
    __shared__ __align__(16) float sNK[NOBJ_PAD];
    __shared__ __align__(16) float sATT[NOBJ_PAD];
    __shared__ __align__(16) float sREP[NOBJ_PAD];
    __shared__ float redA[256];
    __shared__ float redB[256];

    const int tid = threadIdx.x;
    for (int j = tid; j < NOBJ_PAD; j += 256) {
        sXKX[j] = obj[j];
        sXKY[j] = obj[544 + j];
        sXKZ[j] = obj[1088 + j];
        sNK[j]  = obj[1632 + j];
        sATT[j] = obj[2176 + j];
        sREP[j] = obj[2720 + j];
    }
    __syncthreads();

    const int wv = tid >> 5;
    const int lane = tid & 31;
    const bool lo = lane < 16;
    const int cidx = lane & 15;
    const int tileI = blockIdx.x * 8 + wv;
    const int nTilesI = (N + 15) >> 4;
    const int Kv = *pK;
    const int nTilesK = (Kv + 1 + 15) >> 4;   // 33 for K=512

    float acc = 0.f;
    float bcol = 0.f;
    float noiseB = 0.f;

    if (tileI < nTilesI) {                    // wave-uniform: EXEC stays all-ones inside
        const int col = tileI * 16 + cidx;
        const int cc = col < N ? col : N - 1;
        const float x = ccoords[3 * cc + 0];
        const float y = ccoords[3 * cc + 1];
        const float z = ccoords[3 * cc + 2];
        const float b = beta[cc];
        const int vid = vids[cc];
        const bool valid = col < N;
        bcol = valid ? b : 0.f;
        const int vidp1 = valid ? (vid + 1) : 0;   // clamped; bcol=0 kills invalid columns
        noiseB = (lo && valid && vid == -1) ? b : 0.f;  // counted once per column (lo half)
        const float normi = x * x + y * y + z * z;

        // B-matrix (4x16, K x N): lanes 0-15 hold rows K=0,1 ; lanes 16-31 rows K=2,3
        v2f Bv;
        Bv.x = lo ? (-2.f * x) : (-2.f * z);
        Bv.y = lo ? (-2.f * y) : 1.0f;

        // C accumulator preloaded with ||xi||^2 (same for all 8 rows of this column)
        const v8f cIn = {normi, normi, normi, normi, normi, normi, normi, normi};

        v8f accv = {0.f, 0.f, 0.f, 0.f, 0.f, 0.f, 0.f, 0.f};

        #pragma unroll 2
        for (int kt = 0; kt < nTilesK; ++kt) {
            const int base = kt << 4;
            // A-matrix (16x4, M x K): lanes 0-15: K=0,1 ; lanes 16-31: K=2,3 (M = lane&15)
            v2f Av;
            Av.x = lo ? sXKX[base + cidx] : sXKZ[base + cidx];
            Av.y = lo ? sXKY[base + cidx] : sNK[base + cidx];
            const v8f c = __builtin_amdgcn_wmma_f32_16x16x4_f32(
                    false, Av, false, Bv, (short)0, cIn, false, false);

            // C/D layout: lanes 0-15 rows M=0..7 (vgpr r), lanes 16-31 rows M=8..15
            const int kbase = base + (lo ? 0 : 8);
            const float4 s0 = *(const float4*)(&sREP[kbase]);
            const float4 s1 = *(const float4*)(&sREP[kbase + 4]);
            const float sr[8] = {s0.x, s0.y, s0.z, s0.w, s1.x, s1.y, s1.z, s1.w};
            #pragma unroll
            for (int r = 0; r < 8; ++r) {
                // |c[r]| clamps the tiny negative cancellation; fabs is a free src modifier
                const float dd = __builtin_amdgcn_sqrtf(__builtin_fabsf(c[r]));
                const float t = fmaxf(1.f - dd, 0.f);
                accv[r] = __builtin_fmaf(sr[r], t, accv[r]);
            }
        }

        // ---- member correction, once per column (owned by the lane covering its row) ----
        const int km = vidp1;                         // in [0, Kv] for valid columns
        const float dx = x - sXKX[km];
        const float dy = y - sXKY[km];
        const float dz = z - sXKZ[km];
        const float d2m = dx * dx + dy * dy + dz * dz;    // exact, matches reference d^2
        const float ddm = __builtin_amdgcn_sqrtf(d2m);
        const float corr = sATT[km] * d2m - sREP[km] * fmaxf(1.f - ddm, 0.f);
        const bool ownsCorr = (((km & 8) == 0) == lo);    // row-in-tile <8 -> lo lane owns
        acc = (((accv[0] + accv[1]) + (accv[2] + accv[3])) +
               ((accv[4] + accv[5]) + (accv[6] + accv[7]))) +
              (ownsCorr ? corr : 0.f);
    }

    redA[tid] = acc * bcol;                   // every per-element term scales by b_i
    redB[tid] = noiseB;
    __syncthreads();
    for (int s = 128; s > 0; s >>= 1) {       // deterministic tree reduce
        if (tid < s) { redA[tid] += redA[tid + s]; redB[tid] += redB[tid + s]; }
        __syncthreads();
    }
    if (tid == 0) {
        partials[2 * blockIdx.x]     = redA[0];
        partials[2 * blockIdx.x + 1] = redB[0];
    }
}

// ---------------- kernel 4: deterministic in-order final sum ----------------
__global__ void oc_finish_kernel(const float* __restrict__ partials, int nBlocks,
                                 const float* __restrict__ pLbeta,
                                 const int* __restrict__ pNnoise,
                                 float* __restrict__ out) {
    if (threadIdx.x == 0 && blockIdx.x == 0) {
        float m = 0.f, nz = 0.f;
        for (int i = 0; i < nBlocks; ++i) {
            m  += partials[2 * i];
            nz += partials[2 * i + 1];
        }
        const float Lb = *pLbeta;
        const int nn = *pNnoise;
        const float Ln = (nn > 0) ? (nz / (float)(nn > 1 ? nn : 1)) : 0.f;
        out[0] = m + Lb + Ln;
    }
}

// ---------------- launcher ----------------
extern "C" void kernel_launch(void* const* d_in, const int* in_sizes, int n_in,
                              void* d_out, int out_size, void* d_ws, size_t ws_size,
                              hipStream_t stream) {
    const int*   vids    = (const int*)d_in[0];    // vertex_ids (N,)
    const float* beta    = (const float*)d_in[1];  // beta (N,1)
    const float* ccoords = (const float*)d_in[2];  // ccoords (N,3)
    const int*   pK      = (const int*)d_in[3];    // n_objects scalar on device
    const int N = in_sizes[0];

    // ws layout (bytes):
    char* ws = (char*)d_ws;
    unsigned long long* packed = (unsigned long long*)(ws + 0);       // 544 * 8   = 4352
    int*   nin      = (int*)(ws + 4352);                              // 544 * 4   = 2176
    int*   nnoise   = (int*)(ws + 6528);                              // 4 (+pad)
    float* obj      = (float*)(ws + 6544);                            // 6*544*4   = 13056
    float* pLbeta   = (float*)(ws + 19600);                           // 4 (+pad)
    float* partials = (float*)(ws + 19616);                           // 2 * nBlocks3 * 4

    const int nTilesI = (N + 15) / 16;        // 6250 for N=100000 (exact)
    const int nBlocks3 = (nTilesI + 7) / 8;   // 8 waves (i-tiles) per 256-thread block

    oc_init_kernel<<<(NOBJ_PAD + 255) / 256, 256, 0, stream>>>(packed, nin, nnoise);
    oc_scan_kernel<<<(N + 255) / 256, 256, 0, stream>>>(vids, beta, N, packed, nin, nnoise);
    oc_cond_kernel<<<1, 1024, 0, stream>>>(packed, nin, beta, ccoords, pK, N, obj, pLbeta);
    oc_main_kernel<<<nBlocks3, 256, 0, stream>>>(ccoords, beta, vids, pK, obj, N, partials);
    oc_finish_kernel<<<1, 32, 0, stream>>>(partials, nBlocks3, pLbeta, nnoise, (float*)d_out);
}